// LinkPredictor_45749991637157
// MI455X (gfx1250) — compile-verified
//
#include <hip/hip_runtime.h>
#include <hip/hip_bf16.h>
#include <math.h>

typedef __attribute__((ext_vector_type(2))) float v2f;
typedef __attribute__((ext_vector_type(8))) float v8f;

#define N_NODES 50000
#define N_EDGES 800000
#define N_EGT   200000
#define MT      5   // M-tiles per wave in node GEMMs (3125 tiles = 625*5)
#define CMT     4   // M-tiles per block in classifier (12500 tiles = 3125*4)

// D = A(16x4) * B(4x16) + C, fp32 end-to-end (V_WMMA_F32_16X16X4_F32)
static __device__ __forceinline__ v8f wmma4(v2f a, v2f b, v8f c) {
  return __builtin_amdgcn_wmma_f32_16x16x4_f32(
      /*neg_a=*/false, a, /*neg_b=*/false, b,
      /*c_mod=*/(short)0, c, /*reuse_a=*/false, /*reuse_b=*/false);
}

__global__ void k_zero(float* __restrict__ p, int count) {
  int i = blockIdx.x * blockDim.x + threadIdx.x;
  if (i < count) p[i] = 0.0f;
}

// h0[:, 0:16] = x
__global__ void k_copy_x(const float* __restrict__ x, float* __restrict__ h0) {
  int i = blockIdx.x * blockDim.x + threadIdx.x;
  if (i < N_NODES * 16) {
    int node = i >> 4, j = i & 15;
    h0[node * 24 + j] = x[i];
  }
}

// scatter edge embeddings into h0 cols [16:20] (trigger->src) and [20:24] (action->dst); count deg[dst]
__global__ void k_edge_emb(const int* __restrict__ ei, const int* __restrict__ ef,
                           const float* __restrict__ embT, const float* __restrict__ embA,
                           float* __restrict__ h0, float* __restrict__ deg) {
  int e = blockIdx.x * blockDim.x + threadIdx.x;
  if (e >= N_EDGES) return;
  int s = ei[e], d = ei[N_EDGES + e];
  int t = ef[2 * e], a = ef[2 * e + 1];
#pragma unroll
  for (int j = 0; j < 4; ++j) {
    atomicAdd(&h0[s * 24 + 16 + j], embT[t * 4 + j]);
    atomicAdd(&h0[d * 24 + 20 + j], embA[a * 4 + j]);
  }
  atomicAdd(&deg[d], 1.0f);
}

// deg -> 1/max(deg,1), in place
__global__ void k_invdeg(float* __restrict__ deg) {
  int i = blockIdx.x * blockDim.x + threadIdx.x;
  if (i < N_NODES) deg[i] = 1.0f / fmaxf(deg[i], 1.0f);
}

// agg1[dst] += h0[src], 24-wide
__global__ void k_agg1(const int* __restrict__ ei, const float* __restrict__ h0,
                       float* __restrict__ agg1) {
  int i = blockIdx.x * blockDim.x + threadIdx.x;
  if (i >= N_EDGES * 24) return;
  int e = i / 24, j = i - e * 24;
  int s = ei[e], d = ei[N_EDGES + e];
  atomicAdd(&agg1[d * 24 + j], h0[s * 24 + j]);
}

// agg2[dst] += p1[src], 128-wide, float4 gather per thread
__global__ void k_agg2(const int* __restrict__ ei, const float* __restrict__ p1,
                       float* __restrict__ agg2) {
  int i = blockIdx.x * blockDim.x + threadIdx.x;
  if (i >= N_EDGES * 32) return;
  int e = i >> 5, q = (i & 31) << 2;
  int s = ei[e], d = ei[N_EDGES + e];
  const float4 v = *(const float4*)(p1 + (size_t)s * 128 + q);
  float* base = agg2 + (size_t)d * 128 + q;
  atomicAdd(base + 0, v.x);
  atomicAdd(base + 1, v.y);
  atomicAdd(base + 2, v.z);
  atomicAdd(base + 3, v.w);
}

// h1 = relu(agg1*invdeg @ W1l + h0 @ W1r + b1)   [N x 24] x [24 x 256]
// 5 M-tiles per wave: one B load feeds 5 WMMAs.
__global__ __launch_bounds__(32) void k_gemm1(
    const float* __restrict__ agg1, const float* __restrict__ h0,
    const float* __restrict__ invdeg,
    const float* __restrict__ W1l, const float* __restrict__ W1r,
    const float* __restrict__ b1, float* __restrict__ h1) {
  int lane = threadIdx.x;
  int l15 = lane & 15, hi = lane >> 4;
  int n = blockIdx.y * 16 + l15;
  int m0 = blockIdx.x * (MT * 16);
  const float* arow[MT];
  const float* hrow[MT];
  float id[MT];
  v8f acc[MT];
  const v8f vzero = {};
#pragma unroll
  for (int i = 0; i < MT; ++i) {
    int m = m0 + i * 16 + l15;
    arow[i] = agg1 + m * 24;
    hrow[i] = h0 + m * 24;
    id[i] = invdeg[m];
    acc[i] = vzero;
  }
#pragma unroll
  for (int kk = 0; kk < 24; kk += 4) {
    int ko = kk + 2 * hi;
    v2f b; b.x = W1l[ko * 256 + n]; b.y = W1l[(ko + 1) * 256 + n];
#pragma unroll
    for (int i = 0; i < MT; ++i) {
      v2f a = *(const v2f*)(arow[i] + ko);
      a.x *= id[i]; a.y *= id[i];
      acc[i] = wmma4(a, b, acc[i]);
    }
  }
#pragma unroll
  for (int kk = 0; kk < 24; kk += 4) {
    int ko = kk + 2 * hi;
    v2f b; b.x = W1r[ko * 256 + n]; b.y = W1r[(ko + 1) * 256 + n];
#pragma unroll
    for (int i = 0; i < MT; ++i) {
      v2f a = *(const v2f*)(hrow[i] + ko);
      acc[i] = wmma4(a, b, acc[i]);
    }
  }
  float bias = b1[n];
#pragma unroll
  for (int i = 0; i < MT; ++i) {
    int mb = m0 + i * 16 + (hi << 3);
#pragma unroll
    for (int r = 0; r < 8; ++r)
      h1[(size_t)(mb + r) * 256 + n] = fmaxf(acc[i][r] + bias, 0.0f);
  }
}

// p1 = h1 @ W2l   [N x 256] x [256 x 128], 5 M-tiles per wave
__global__ __launch_bounds__(32) void k_gemm_p1(
    const float* __restrict__ h1, const float* __restrict__ W2l,
    float* __restrict__ p1) {
  int lane = threadIdx.x;
  int l15 = lane & 15, hi = lane >> 4;
  int n = blockIdx.y * 16 + l15;
  int m0 = blockIdx.x * (MT * 16);
  const float* arow[MT];
  v8f acc[MT];
  const v8f vzero = {};
#pragma unroll
  for (int i = 0; i < MT; ++i) {
    arow[i] = h1 + (size_t)(m0 + i * 16 + l15) * 256;
    acc[i] = vzero;
  }
#pragma unroll 4
  for (int kk = 0; kk < 256; kk += 4) {
    int ko = kk + 2 * hi;
    v2f b; b.x = W2l[ko * 128 + n]; b.y = W2l[(ko + 1) * 128 + n];
#pragma unroll
    for (int i = 0; i < MT; ++i) {
      v2f a = *(const v2f*)(arow[i] + ko);
      acc[i] = wmma4(a, b, acc[i]);
    }
  }
#pragma unroll
  for (int i = 0; i < MT; ++i) {
    int mb = m0 + i * 16 + (hi << 3);
#pragma unroll
    for (int r = 0; r < 8; ++r)
      p1[(size_t)(mb + r) * 128 + n] = acc[i][r];
  }
}

// h2 = relu(h1 @ W2r + agg2*invdeg + b2), in place into agg2; 5 M-tiles per wave
__global__ __launch_bounds__(32) void k_gemm_h2(
    const float* __restrict__ h1, const float* __restrict__ W2r,
    const float* __restrict__ b2, const float* __restrict__ invdeg,
    float* __restrict__ agg2) {
  int lane = threadIdx.x;
  int l15 = lane & 15, hi = lane >> 4;
  int n = blockIdx.y * 16 + l15;
  int m0 = blockIdx.x * (MT * 16);
  const float* arow[MT];
  v8f acc[MT];
  const v8f vzero = {};
#pragma unroll
  for (int i = 0; i < MT; ++i) {
    arow[i] = h1 + (size_t)(m0 + i * 16 + l15) * 256;
    acc[i] = vzero;
  }
#pragma unroll 4
  for (int kk = 0; kk < 256; kk += 4) {
    int ko = kk + 2 * hi;
    v2f b; b.x = W2r[ko * 128 + n]; b.y = W2r[(ko + 1) * 128 + n];
#pragma unroll
    for (int i = 0; i < MT; ++i) {
      v2f a = *(const v2f*)(arow[i] + ko);
      acc[i] = wmma4(a, b, acc[i]);
    }
  }
  float bias = b2[n];
#pragma unroll
  for (int i = 0; i < MT; ++i) {
    int mb = m0 + i * 16 + (hi << 3);
#pragma unroll
    for (int r = 0; r < 8; ++r) {
      size_t idx = (size_t)(mb + r) * 128 + n;
      float v = acc[i][r] + agg2[idx] * invdeg[mb + r] + bias;
      agg2[idx] = fmaxf(v, 0.0f);
    }
  }
}

// fused classifier, 4 edge-tiles (64 edges) per block:
// z = relu([h2[s]|h2[d]] @ Wc1 + bc1) staged in LDS; out = sigmoid(z @ Wc2 + bc2)
__global__ __launch_bounds__(256) void k_classify(
    const float* __restrict__ h2, const int* __restrict__ gt,
    const float* __restrict__ Wc1, const float* __restrict__ bc1,
    const float* __restrict__ Wc2, const float* __restrict__ bc2,
    float* __restrict__ out) {
  __shared__ float zs[16 * CMT][132];  // 64 rows x 128 cols, padded vs 64 banks
  int lane = threadIdx.x & 31;
  int wave = threadIdx.x >> 5;     // 0..7
  int l15 = lane & 15, hi = lane >> 4;
  const float* rowS[CMT];
  const float* rowD[CMT];
#pragma unroll
  for (int i = 0; i < CMT; ++i) {
    int m = blockIdx.x * (16 * CMT) + i * 16 + l15;
    rowS[i] = h2 + (size_t)gt[m] * 128;
    rowD[i] = h2 + (size_t)gt[N_EGT + m] * 128;
  }
  const v8f vzero = {};

  // phase 1: wave w computes z columns [w*16, w*16+16) for all 4 edge-tiles
  {
    int n = wave * 16 + l15;
    v8f acc[CMT];
#pragma unroll
    for (int i = 0; i < CMT; ++i) acc[i] = vzero;
#pragma unroll 2
    for (int kk = 0; kk < 256; kk += 4) {
      int ko = kk + 2 * hi;
      v2f b; b.x = Wc1[ko * 128 + n]; b.y = Wc1[(ko + 1) * 128 + n];
#pragma unroll
      for (int i = 0; i < CMT; ++i) {
        const float* ap = (ko < 128) ? (rowS[i] + ko) : (rowD[i] + (ko - 128));
        v2f a = *(const v2f*)ap;
        acc[i] = wmma4(a, b, acc[i]);
      }
    }
    float bias = bc1[n];
#pragma unroll
    for (int i = 0; i < CMT; ++i)
#pragma unroll
      for (int r = 0; r < 8; ++r)
        zs[i * 16 + (hi << 3) + r][n] = fmaxf(acc[i][r] + bias, 0.0f);
  }
  __syncthreads();

  // phase 2: 35 output n-tiles (552 cols, last partial), round-robin over waves;
  // each B load feeds 4 WMMAs (one per edge-tile)
  for (int t = wave; t < 35; t += 8) {
    int nn = t * 16 + l15;
    bool nok = nn < 552;
    v8f acc[CMT];
#pragma unroll
    for (int i = 0; i < CMT; ++i) acc[i] = vzero;
#pragma unroll 2
    for (int kk = 0; kk < 128; kk += 4) {
      int ko = kk + 2 * hi;
      v2f b;
      b.x = nok ? Wc2[ko * 552 + nn] : 0.0f;
      b.y = nok ? Wc2[(ko + 1) * 552 + nn] : 0.0f;
#pragma unroll
      for (int i = 0; i < CMT; ++i) {
        v2f a = *(const v2f*)(&zs[i * 16 + l15][ko]);
        acc[i] = wmma4(a, b, acc[i]);  // full-EXEC: guard on data, not lanes
      }
    }
    if (nok) {
      float bias = bc2[nn];
#pragma unroll
      for (int i = 0; i < CMT; ++i) {
        int mb = blockIdx.x * (16 * CMT) + i * 16 + (hi << 3);
#pragma unroll
        for (int r = 0; r < 8; ++r) {
          float v = acc[i][r] + bias;
          out[(size_t)(mb + r) * 552 + nn] = 1.0f / (1.0f + __expf(-v));
        }
      }
    }
  }
}

extern "C" void kernel_launch(void* const* d_in, const int* in_sizes, int n_in,
                              void* d_out, int out_size, void* d_ws, size_t ws_size,
                              hipStream_t stream) {
  (void)in_sizes; (void)n_in; (void)out_size; (void)ws_size;
  const float* x    = (const float*)d_in[0];
  const int*   ei   = (const int*)d_in[1];   // (2,E)
  const int*   gt   = (const int*)d_in[2];   // (2,EGT)
  const int*   ef   = (const int*)d_in[3];   // (E,2)
  const float* embT = (const float*)d_in[4];
  const float* embA = (const float*)d_in[5];
  const float* W1l  = (const float*)d_in[6];
  const float* W1r  = (const float*)d_in[7];
  const float* b1   = (const float*)d_in[8];
  const float* W2l  = (const float*)d_in[9];
  const float* W2r  = (const float*)d_in[10];
  const float* b2   = (const float*)d_in[11];
  const float* Wc1  = (const float*)d_in[12];
  const float* bc1  = (const float*)d_in[13];
  const float* Wc2  = (const float*)d_in[14];
  const float* bc2  = (const float*)d_in[15];
  float* out = (float*)d_out;

  // workspace layout (floats): deg | h0(N*24) | agg1(N*24) | h1(N*256) | p1(N*128) | agg2/h2(N*128)
  float* ws   = (float*)d_ws;
  float* deg  = ws;                              // -> invdeg after k_invdeg
  float* h0   = deg + (size_t)N_NODES;
  float* agg1 = h0 + (size_t)N_NODES * 24;
  float* h1   = agg1 + (size_t)N_NODES * 24;
  float* p1   = h1 + (size_t)N_NODES * 256;
  float* agg2 = p1 + (size_t)N_NODES * 128;

  const int T = 256;
  k_zero<<<(N_NODES * 49 + T - 1) / T, T, 0, stream>>>(deg, N_NODES * 49);       // deg+h0+agg1
  k_zero<<<(N_NODES * 128 + T - 1) / T, T, 0, stream>>>(agg2, N_NODES * 128);
  k_copy_x<<<(N_NODES * 16 + T - 1) / T, T, 0, stream>>>(x, h0);
  k_edge_emb<<<(N_EDGES + T - 1) / T, T, 0, stream>>>(ei, ef, embT, embA, h0, deg);
  k_invdeg<<<(N_NODES + T - 1) / T, T, 0, stream>>>(deg);
  k_agg1<<<(N_EDGES * 24 + T - 1) / T, T, 0, stream>>>(ei, h0, agg1);
  k_gemm1<<<dim3(N_NODES / (16 * MT), 256 / 16), 32, 0, stream>>>(agg1, h0, deg, W1l, W1r, b1, h1);
  k_gemm_p1<<<dim3(N_NODES / (16 * MT), 128 / 16), 32, 0, stream>>>(h1, W2l, p1);
  k_agg2<<<(N_EDGES * 32 + T - 1) / T, T, 0, stream>>>(ei, p1, agg2);
  k_gemm_h2<<<dim3(N_NODES / (16 * MT), 128 / 16), 32, 0, stream>>>(h1, W2r, b2, deg, agg2);
  k_classify<<<N_EGT / (16 * CMT), 256, 0, stream>>>(agg2, gt, Wc1, bc1, Wc2, bc2, out);
}